// MoEGate_72138270703850
// MI455X (gfx1250) — compile-verified
//
#include <hip/hip_runtime.h>
#include <stdint.h>

// ---------------------------------------------------------------------------
// MoE gate for MI455X (gfx1250, wave32):
//   logits = x @ W^T  (fp32, via V_WMMA_F32_16X16X4_F32)
//   softmax over 64 experts, top-8 weights + indices.
// Memory-bound: ~336 MB traffic -> ~14us floor @ 23.3 TB/s.
// ---------------------------------------------------------------------------

typedef __attribute__((ext_vector_type(2))) float v2f;
typedef __attribute__((ext_vector_type(8))) float v8f;

#define N_EXPERTS        64
#define N_TOPK           8
#define THREADS          128          // 4 wave32
#define TOKENS_PER_BLOCK 128
#define ROW_STRIDE       68           // padded floats per token row in LDS

// Monotone float -> uint map (IEEE order-preserving).
__device__ __forceinline__ unsigned ordf(float f) {
    unsigned u = __float_as_uint(f);
    return (u & 0x80000000u) ? ~u : (u | 0x80000000u);
}

// Sortable key: top 26 bits of ordered float, low 6 bits = (63 - expert)
// so that ties prefer the LOWER expert index (matches jax.lax.top_k).
__device__ __forceinline__ unsigned make_key(float v, int e) {
    return (ordf(v) & 0xFFFFFFC0u) | (unsigned)(63 - e);
}

__global__ __launch_bounds__(THREADS)
void moe_gate_kernel(const float* __restrict__ x,
                     const float* __restrict__ w,
                     float* __restrict__ out_w,
                     int*   __restrict__ out_i,
                     int n_tokens) {
    // B-operand table: [kc(16)][nb(4)][lane(32)] float2  -> 16 KB
    __shared__ v2f   btab[16 * 4 * 32];
    // Logits staging: [token(128)][expert(64)] padded    -> 34 KB
    __shared__ float logits[TOKENS_PER_BLOCK * ROW_STRIDE];

    const int tid  = threadIdx.x;
    const int lane = tid & 31;
    const int wave = tid >> 5;      // 0..3
    const int l15  = lane & 15;
    const int lhi  = lane >> 4;     // 0 or 1

    // ---- Stage W^T into WMMA B-matrix layout (ISA 7.12.2):
    //   B[k][n], k = v + 2*(lane>=16), n = lane&15
    //   B[e][f] = W[f*64 + e]  (f = 16*nb + n, e = 4*kc + k)
    for (int q = tid; q < 16 * 4 * 32; q += THREADS) {
        int ql  = q & 31;
        int qnb = (q >> 5) & 3;
        int qkc = q >> 7;
        int f   = 16 * qnb + (ql & 15);
        int e   = 4 * qkc + 2 * (ql >> 4);
        v2f b;
        b.x = w[f * N_EXPERTS + e];
        b.y = w[f * N_EXPERTS + e + 1];
        btab[q] = b;
    }
    __syncthreads();

    // ---- Phase 1: GEMM via v_wmma_f32_16x16x4_f32 -----------------------
    const int block_tok0 = blockIdx.x * TOKENS_PER_BLOCK;

    for (int tile = 0; tile < 2; ++tile) {
        const int     ltok0 = wave * 32 + tile * 16;       // local token base
        const int64_t gtok  = (int64_t)(block_tok0 + ltok0 + l15);
        const float*  row   = x + gtok * N_EXPERTS;

        // A-matrix 16x4 tiles (ISA layout: m=lane&15, k = v + 2*(lane>=16)).
        // Lane reads 16 contiguous float2 chunks of its token's 256B row.
        v2f a[16];
#pragma unroll
        for (int kc = 0; kc < 16; ++kc)
            a[kc] = *(const v2f*)(row + 4 * kc + 2 * lhi);

        v8f c0 = {}, c1 = {}, c2 = {}, c3 = {};
#pragma unroll
        for (int kc = 0; kc < 16; ++kc) {
            const v2f* bt = &btab[kc * 128 + lane];
            c0 = __builtin_amdgcn_wmma_f32_16x16x4_f32(false, a[kc], false, bt[0],
                                                       (short)0, c0, false, false);
            c1 = __builtin_amdgcn_wmma_f32_16x16x4_f32(false, a[kc], false, bt[32],
                                                       (short)0, c1, false, false);
            c2 = __builtin_amdgcn_wmma_f32_16x16x4_f32(false, a[kc], false, bt[64],
                                                       (short)0, c2, false, false);
            c3 = __builtin_amdgcn_wmma_f32_16x16x4_f32(false, a[kc], false, bt[96],
                                                       (short)0, c3, false, false);
        }

        // C layout: vgpr r, lanes0-15 -> M=r, lanes16-31 -> M=r+8; N=lane&15.
        // Scatter to logits[token][expert]; ROW_STRIDE=68 keeps the two lane
        // halves on disjoint bank sets (8*68 mod 64 == 32).
        float* lrow = logits + (ltok0 + 8 * lhi) * ROW_STRIDE + l15;
#pragma unroll
        for (int r = 0; r < 8; ++r) {
            lrow[r * ROW_STRIDE +  0] = c0[r];
            lrow[r * ROW_STRIDE + 16] = c1[r];
            lrow[r * ROW_STRIDE + 32] = c2[r];
            lrow[r * ROW_STRIDE + 48] = c3[r];
        }
    }
    __syncthreads();

    // ---- Phase 2: softmax + top-8, one token per thread ------------------
    const int    tok = block_tok0 + tid;
    const float* lr  = logits + tid * ROW_STRIDE;

    // Pass A: build sortable keys + running max.
    unsigned key[N_EXPERTS];
    float    m = -3.402823466e38f;
#pragma unroll
    for (int e = 0; e < N_EXPERTS; e += 4) {
        float4 t4 = *(const float4*)(lr + e);
        m = fmaxf(m, fmaxf(fmaxf(t4.x, t4.y), fmaxf(t4.z, t4.w)));
        key[e + 0] = make_key(t4.x, e + 0);
        key[e + 1] = make_key(t4.y, e + 1);
        key[e + 2] = make_key(t4.z, e + 2);
        key[e + 3] = make_key(t4.w, e + 3);
    }

    // Pass B: softmax denominator (exact fp32 logits re-read from LDS).
    float s = 0.0f;
#pragma unroll
    for (int e = 0; e < N_EXPERTS; e += 4) {
        float4 t4 = *(const float4*)(lr + e);
        s += __expf(t4.x - m) + __expf(t4.y - m) +
             __expf(t4.z - m) + __expf(t4.w - m);
    }
    const float inv = 1.0f / s;

    // 8 branch-free argmax + knockout passes.
    float wv[N_TOPK];
    int   iv[N_TOPK];
#pragma unroll
    for (int j = 0; j < N_TOPK; ++j) {
        unsigned best = 0u;
#pragma unroll
        for (int e = 0; e < N_EXPERTS; ++e)
            best = (key[e] > best) ? key[e] : best;
        const int idx = 63 - (int)(best & 63u);
        wv[j] = __expf(lr[idx] - m) * inv;   // exact fp32 logit from LDS
        iv[j] = idx;
#pragma unroll
        for (int e = 0; e < N_EXPERTS; ++e)
            key[e] = (key[e] == best) ? 0u : key[e];
    }

    if (tok < n_tokens) {
        float* dw = out_w + (size_t)tok * N_TOPK;
        int*   di = out_i + (size_t)tok * N_TOPK;
        *(float4*)(dw)     = make_float4(wv[0], wv[1], wv[2], wv[3]);
        *(float4*)(dw + 4) = make_float4(wv[4], wv[5], wv[6], wv[7]);
        *(int4*)(di)       = make_int4(iv[0], iv[1], iv[2], iv[3]);
        *(int4*)(di + 4)   = make_int4(iv[4], iv[5], iv[6], iv[7]);
    }
}

extern "C" void kernel_launch(void* const* d_in, const int* in_sizes, int n_in,
                              void* d_out, int out_size, void* d_ws, size_t ws_size,
                              hipStream_t stream) {
    const float* x = (const float*)d_in[0];   // [n_tokens, 64] fp32
    const float* w = (const float*)d_in[1];   // [64, 64] fp32

    const int n_tokens = in_sizes[0] / N_EXPERTS;

    float* out_w = (float*)d_out;                                   // [n_tokens, 8] fp32
    int*   out_i = (int*)((float*)d_out + (size_t)n_tokens * N_TOPK); // [n_tokens, 8] i32

    const int blocks = (n_tokens + TOKENS_PER_BLOCK - 1) / TOKENS_PER_BLOCK;
    hipLaunchKernelGGL(moe_gate_kernel, dim3(blocks), dim3(THREADS), 0, stream,
                       x, w, out_w, out_i, n_tokens);
}